// LossMOEDecoder_1108101562793
// MI455X (gfx1250) — compile-verified
//
#include <hip/hip_runtime.h>
#include <hip/hip_bf16.h>
#include <cstddef>

#define NTOK 256   // B*S tokens
#define DIM  768
#define HID  2048
#define BOT  256
#define OUTD 768
#define NEXP 32
#define MAXSLOT 256  // worst case: every token routed to one expert

typedef float v2f __attribute__((ext_vector_type(2)));
typedef float v8f __attribute__((ext_vector_type(8)));

// ---------------------------------------------------------------------------
// Kernel 1: gate + routing.  One wave (32 lanes) per token; lane e computes
// the logit for expert e, then softmax across the wave, top-2, normalized
// weights, and appends (token, weight) to that expert's list via atomics.
// ---------------------------------------------------------------------------
__global__ __launch_bounds__(32) void gate_route_kernel(
    const float* __restrict__ x, const float* __restrict__ gw,
    int* __restrict__ counts, int* __restrict__ toks, float* __restrict__ wvals)
{
    const int t = blockIdx.x;      // token
    const int e = threadIdx.x;     // expert (one per lane)
    const float* xr = x + (size_t)t * DIM;
    const float* wr = gw + (size_t)e * DIM;
    float acc = 0.0f;
    #pragma unroll 4
    for (int d = 0; d < DIM; ++d) acc += xr[d] * wr[d];

    __shared__ float s[NEXP];
    s[e] = acc;
    __syncthreads();

    if (e == 0) {
        float mx = s[0];
        for (int i = 1; i < NEXP; ++i) mx = fmaxf(mx, s[i]);
        float sc[NEXP];
        float sum = 0.0f;
        for (int i = 0; i < NEXP; ++i) { sc[i] = __expf(s[i] - mx); sum += sc[i]; }
        int i1 = 0;
        for (int i = 1; i < NEXP; ++i) if (sc[i] > sc[i1]) i1 = i;
        int i2 = (i1 == 0) ? 1 : 0;
        for (int i = 0; i < NEXP; ++i) if (i != i1 && sc[i] > sc[i2]) i2 = i;
        float w1 = sc[i1] / sum, w2 = sc[i2] / sum;
        float norm = w1 + w2 + 1e-20f;
        w1 /= norm; w2 /= norm;
        int s1 = atomicAdd(&counts[i1], 1);
        toks[i1 * MAXSLOT + s1] = t;  wvals[i1 * MAXSLOT + s1] = w1;
        int s2 = atomicAdd(&counts[i2], 1);
        toks[i2 * MAXSLOT + s2] = t;  wvals[i2 * MAXSLOT + s2] = w2;
    }
}

// ---------------------------------------------------------------------------
// WMMA helpers (fp32 16x16x4).
//   A (16x4):  lanes 0-15 row=lane, K=k0/k0+1; lanes 16-31 row=lane-16, K=k0+2/k0+3
//   B (4x16):  lanes 0-15 col=lane, K=k0/k0+1; lanes 16-31 col=lane-16, K=k0+2/k0+3
//   C/D:       vgpr v, lane L -> row = v + 8*(L>=16), col = L%16
// ---------------------------------------------------------------------------
__device__ __forceinline__ v8f wmma_gemm_tile(const float* __restrict__ Alds, int K,
                                              const float* __restrict__ Wg, int Nw, int n0)
{
    const int lane  = threadIdx.x & 31;
    const int r     = lane & 15;            // A row / B col within tile
    const int khalf = (lane >> 4) << 1;     // 0 or 2
    v8f acc = {};
    for (int k0 = 0; k0 < K; k0 += 4) {
        const int ka = k0 + khalf;
        v2f a;
        a.x = Alds[r * K + ka];
        a.y = Alds[r * K + ka + 1];
        const float* wp = Wg + (size_t)ka * Nw + n0 + r;
        v2f b;
        b.x = wp[0];
        b.y = wp[Nw];
        acc = __builtin_amdgcn_wmma_f32_16x16x4_f32(
                  false, a, false, b, (short)0, acc, false, false);
    }
    return acc;
}

__device__ __forceinline__ void store_tile(v8f acc, const float* __restrict__ bias,
                                           int n0, float* __restrict__ Out, int Nw,
                                           bool gelu)
{
    const int lane  = threadIdx.x & 31;
    const int c     = lane & 15;
    const int rbase = (lane >> 4) * 8;
    const float bv  = bias[n0 + c];
    #pragma unroll
    for (int v = 0; v < 8; ++v) {
        float val = acc[v] + bv;
        if (gelu) val = 0.5f * val * (1.0f + erff(val * 0.70710678118654752f));
        Out[(rbase + v) * Nw + n0 + c] = val;
    }
}

// ---------------------------------------------------------------------------
// Kernel 2: expert MLP. grid = NEXP * 16 tiles, block = 256 threads (8 waves).
// LDS: regionA/regionB of 16x2048 fp32 each (ping-pong activations).
// ---------------------------------------------------------------------------
__global__ __launch_bounds__(256) void moe_expert_kernel(
    const float* __restrict__ x,
    const float* __restrict__ W1, const float* __restrict__ b1,
    const float* __restrict__ W2, const float* __restrict__ b2,
    const float* __restrict__ W3, const float* __restrict__ b3,
    const float* __restrict__ Wl, const float* __restrict__ bl,
    const int* __restrict__ counts, const int* __restrict__ toks,
    const float* __restrict__ wvals, float* __restrict__ out)
{
    extern __shared__ float lds[];
    float* regionA = lds;              // 32768 floats (128 KB)
    float* regionB = lds + 16 * HID;   // 32768 floats (128 KB)
    __shared__ int   s_tok[16];
    __shared__ float s_w[16];

    const int e    = blockIdx.x >> 4;
    const int tile = blockIdx.x & 15;
    const int cnt  = counts[e];
    const int base = tile * 16;
    if (base >= cnt) return;
    const int mvalid = min(16, cnt - base);

    const int tid = threadIdx.x;
    if (tid < 16) {
        const bool v = tid < mvalid;
        s_tok[tid] = v ? toks[e * MAXSLOT + base + tid] : 0;
        s_w[tid]   = v ? wvals[e * MAXSLOT + base + tid] : 0.0f;
    }
    __syncthreads();

    // Stage X tile (16 x 768) into regionA, zero-padding invalid rows.
    for (int i = tid; i < 16 * DIM; i += 256) {
        const int r = i / DIM, c = i % DIM;
        regionA[r * DIM + c] = (r < mvalid) ? x[(size_t)s_tok[r] * DIM + c] : 0.0f;
    }
    __syncthreads();

    const int wave = tid >> 5;  // 0..7

    // Layer 1: X(16x768) @ W1 -> GELU -> H1 in regionB (16x2048)
    const float* W1e = W1 + (size_t)e * DIM * HID;
    for (int tn = wave; tn < HID / 16; tn += 8) {
        v8f acc = wmma_gemm_tile(regionA, DIM, W1e, HID, tn * 16);
        store_tile(acc, b1 + (size_t)e * HID, tn * 16, regionB, HID, true);
    }
    __syncthreads();

    // Layer 2: H1(16x2048) @ W2 -> GELU -> H2 in regionA (16x2048)
    const float* W2e = W2 + (size_t)e * HID * HID;
    for (int tn = wave; tn < HID / 16; tn += 8) {
        v8f acc = wmma_gemm_tile(regionB, HID, W2e, HID, tn * 16);
        store_tile(acc, b2 + (size_t)e * HID, tn * 16, regionA, HID, true);
    }
    __syncthreads();

    // Layer 3: H2(16x2048) @ W3 -> Z in regionB (16x256), no activation
    const float* W3e = W3 + (size_t)e * HID * BOT;
    for (int tn = wave; tn < BOT / 16; tn += 8) {
        v8f acc = wmma_gemm_tile(regionA, HID, W3e, BOT, tn * 16);
        store_tile(acc, b3 + (size_t)e * BOT, tn * 16, regionB, BOT, false);
    }
    __syncthreads();

    // Layer 4: Z(16x256) @ Wl + bl, scale by gate weight, scatter-add to out.
    const float* Wle  = Wl + (size_t)e * BOT * OUTD;
    const int lane    = tid & 31;
    const int c       = lane & 15;
    const int rbase   = (lane >> 4) * 8;
    for (int tn = wave; tn < OUTD / 16; tn += 8) {
        v8f acc = wmma_gemm_tile(regionB, BOT, Wle, OUTD, tn * 16);
        const float bv = bl[(size_t)e * OUTD + tn * 16 + c];
        #pragma unroll
        for (int v = 0; v < 8; ++v) {
            const int r = rbase + v;
            if (r < mvalid) {
                const float val = (acc[v] + bv) * s_w[r];
                unsafeAtomicAdd(&out[(size_t)s_tok[r] * OUTD + tn * 16 + c], val);
            }
        }
    }
}

// ---------------------------------------------------------------------------
extern "C" void kernel_launch(void* const* d_in, const int* in_sizes, int n_in,
                              void* d_out, int out_size, void* d_ws, size_t ws_size,
                              hipStream_t stream)
{
    (void)in_sizes; (void)n_in; (void)ws_size;
    const float* x  = (const float*)d_in[0];
    const float* gw = (const float*)d_in[1];
    const float* W1 = (const float*)d_in[2];
    const float* b1 = (const float*)d_in[3];
    const float* W2 = (const float*)d_in[4];
    const float* b2 = (const float*)d_in[5];
    const float* W3 = (const float*)d_in[6];
    const float* b3 = (const float*)d_in[7];
    const float* Wl = (const float*)d_in[8];
    const float* bl = (const float*)d_in[9];
    float* out = (float*)d_out;

    int*   counts = (int*)d_ws;
    int*   toks   = (int*)((char*)d_ws + 128);
    float* wvals  = (float*)((char*)d_ws + 128 + NEXP * MAXSLOT * sizeof(int));

    hipMemsetAsync(counts, 0, NEXP * sizeof(int), stream);
    hipMemsetAsync(d_out, 0, (size_t)out_size * sizeof(float), stream);

    gate_route_kernel<<<NTOK, 32, 0, stream>>>(x, gw, counts, toks, wvals);

    const int lds_bytes = 2 * 16 * HID * sizeof(float);  // 256 KB
    hipFuncSetAttribute((const void*)moe_expert_kernel,
                        hipFuncAttributeMaxDynamicSharedMemorySize, lds_bytes);
    moe_expert_kernel<<<dim3(NEXP * 16), dim3(256), lds_bytes, stream>>>(
        x, W1, b1, W2, b2, W3, b3, Wl, bl, counts, toks, wvals, out);
}